// DiffMambaNet_68676527063210
// MI455X (gfx1250) — compile-verified
//
#include <hip/hip_runtime.h>
#include <hip/hip_bf16.h>
#include <math.h>

#define NB   4
#define NC   64
#define NP   8192
#define NO   128
#define NK   16
#define BN_EPSF 1e-5f
#define SLOPE 0.2f

typedef __attribute__((ext_vector_type(2))) float v2f;
typedef __attribute__((ext_vector_type(8))) float v8f;
typedef __attribute__((ext_vector_type(4))) unsigned int v4u;
typedef __attribute__((ext_vector_type(4))) int v4i;
typedef __attribute__((ext_vector_type(8))) int v8i;

#define BROW 68   // padded LDS row stride in dwords (64 data + 4 pad) -> bank-conflict-free

// ---------------------------------------------------------------------------
// Kernel 1: transpose x (B,C,N) -> xt (B,N,C) and per-point squared norms.
// ---------------------------------------------------------------------------
__global__ __launch_bounds__(256) void k_transpose_sq(const float* __restrict__ x,
                                                      float* __restrict__ xt,
                                                      float* __restrict__ sq)
{
    int id = blockIdx.x * 256 + threadIdx.x;          // 0 .. NB*NP-1
    int b = id / NP;
    int n = id % NP;
    const float* src = x + (size_t)b * NC * NP + n;
    float* dst = xt + ((size_t)b * NP + n) * NC;
    float acc = 0.f;
    #pragma unroll
    for (int c0 = 0; c0 < NC; c0 += 4) {
        float4 v;
        v.x = src[(size_t)(c0 + 0) * NP];
        v.y = src[(size_t)(c0 + 1) * NP];
        v.z = src[(size_t)(c0 + 2) * NP];
        v.w = src[(size_t)(c0 + 3) * NP];
        acc += v.x * v.x + v.y * v.y + v.z * v.z + v.w * v.w;
        *(float4*)(dst + c0) = v;
    }
    sq[id] = acc;
}

// ---------------------------------------------------------------------------
// TDM descriptor: 1-D tile of 128*NC f32 (one column chunk of xt), DMA'd
// Global->LDS with 4-dword padding every 64 dwords (row stride becomes 68).
// ---------------------------------------------------------------------------
__device__ __forceinline__ void tdm_load_chunk(const float* gsrc, unsigned int lds_byte_off)
{
    unsigned long long ga = (unsigned long long)(uintptr_t)gsrc;
    v4u g0;
    g0.x = 1u;                                    // count=1, user mode
    g0.y = lds_byte_off;                          // lds_addr
    g0.z = (unsigned int)ga;                      // global_addr[31:0]
    g0.w = (unsigned int)(ga >> 32) | (2u << 30); // global_addr[56:32] | type=2
    v8i g1;
    // data_size=2 (4B) | pad_enable | pad_interval=5 (64 dw) | pad_amount=3 (4 dw)
    g1[0] = (int)((2u << 16) | (1u << 20) | (5u << 22) | (3u << 25));
    const unsigned int TD0 = 1u << 20;            // tensor_dim0 (large, no OOB clip)
    g1[1] = (int)((TD0 & 0xFFFFu) << 16);         // tensor_dim0[15:0]
    g1[2] = (int)(((TD0 >> 16) & 0xFFFFu) | (1u << 16)); // tensor_dim0[31:16] | tensor_dim1=1
    g1[3] = (int)((unsigned int)(128 * NC) << 16);       // tile_dim0 = 8192 elements
    g1[4] = 0;                                    // tile_dim1=0 (1-D), tile_dim2=0
    g1[5] = (int)(128u * NC);                     // tensor_dim0_stride (unused for 1-D)
    g1[6] = 0;
    g1[7] = 0;
    v4i gz4 = {0, 0, 0, 0};
    v8i gz8 = {0, 0, 0, 0, 0, 0, 0, 0};
    __builtin_amdgcn_tensor_load_to_lds(g0, g1, gz4, gz4, gz8, 0);
}

// ---------------------------------------------------------------------------
// Kernel 2: fused distance GEMM (f32 WMMA 16x16x4) + top-16 selection.
// One WG per (batch, 128-row block); 8 waves own one 16-row tile each.
// Column chunks of 128 points are double-buffered in LDS via the Tensor Data
// Mover (wave 0 issues, TENSORcnt-gated), overlapping DMA with WMMA compute.
// ---------------------------------------------------------------------------
__global__ __launch_bounds__(256) void k_knn(const float* __restrict__ xt,
                                             const float* __restrict__ sq,
                                             int* __restrict__ knn_idx)
{
    __shared__ float ldsB[2][128 * BROW];  // TDM-filled, padded rows (~68 KB)
    __shared__ float ldsSq[2][128];
    __shared__ float ldsD[8][16 * 16];     // per-wave D transpose staging (8 KB)

    const int nblk = NP / 128;
    int b      = blockIdx.x / nblk;
    int row0wg = (blockIdx.x % nblk) * 128;

    int tid  = threadIdx.x;
    int wave = tid >> 5;
    int lane = tid & 31;
    int hs   = lane >> 4;        // which 16-lane half
    int l16  = lane & 15;

    const float* xtb = xt + (size_t)b * NP * NC;
    const float* sqb = sq + (size_t)b * NP;

    unsigned int ldsB_off[2];
    ldsB_off[0] = (unsigned int)(uintptr_t)&ldsB[0][0];
    ldsB_off[1] = (unsigned int)(uintptr_t)&ldsB[1][0];

    int myrow = row0wg + wave * 16 + l16;     // row this lane tracks top-K for
    float sqr = sqb[myrow];

    // A fragments for this wave's 16-row tile (resident in VGPRs all kernel).
    v2f afrag[16];
    {
        const float* arow = xtb + (size_t)myrow * NC;
        #pragma unroll
        for (int kk = 0; kk < 16; ++kk) {
            int k = kk * 4 + hs * 2;
            afrag[kk].x = arow[k];
            afrag[kk].y = arow[k + 1];
        }
    }

    float tv[NK];
    int   ti[NK];
    #pragma unroll
    for (int i = 0; i < NK; ++i) { tv[i] = -__builtin_inff(); ti[i] = 0; }

    if (wave == 0)
        tdm_load_chunk(xtb, ldsB_off[0]);      // prime buffer 0 with chunk 0

    for (int ch = 0; ch < NP / 128; ++ch) {
        int col0 = ch * 128;
        // prefetch next chunk into the other buffer (consumed by the previous chunk),
        // then wait for the current chunk's DMA (TDM ops complete in order).
        if (wave == 0) {
            if (ch + 1 < NP / 128) {
                tdm_load_chunk(xtb + (size_t)(ch + 1) * 128 * NC,
                               ldsB_off[(ch + 1) & 1]);
                __builtin_amdgcn_s_wait_tensorcnt(1);
            } else {
                __builtin_amdgcn_s_wait_tensorcnt(0);
            }
        }
        if (tid < 128) ldsSq[ch & 1][tid] = sqb[col0 + tid];
        __syncthreads();                       // current buffer published
        const float* Bbuf = &ldsB[ch & 1][0];
        const float* Sqc  = &ldsSq[ch & 1][0];

        float* dstD = &ldsD[wave][0];
        for (int ct = 0; ct < 8; ++ct) {
            int c0 = ct * 16;
            const float* brow = Bbuf + (size_t)(c0 + l16) * BROW + hs * 2;
            v8f acc = {};
            #pragma unroll
            for (int kk = 0; kk < 16; ++kk) {
                float2 bp = *(const float2*)(brow + kk * 4);
                v2f bf; bf.x = bp.x; bf.y = bp.y;
                acc = __builtin_amdgcn_wmma_f32_16x16x4_f32(
                        false, afrag[kk], false, bf, (short)0, acc, false, false);
            }
            // Stage D tile to wave-private LDS, row-major [M][Ncol].
            #pragma unroll
            for (int i = 0; i < 8; ++i)
                dstD[(i + hs * 8) * 16 + l16] = acc[i];
            asm volatile("" ::: "memory");     // DS in-order per wave

            // each lane scans 8 candidates of its own row
            #pragma unroll
            for (int j = 0; j < 8; ++j) {
                int cc   = hs * 8 + j;
                int gcol = col0 + c0 + cc;
                float inner = dstD[l16 * 16 + cc];
                float nd = 2.f * inner - sqr - Sqc[c0 + cc];
                nd = (gcol == myrow) ? -__builtin_inff() : nd;
                if (nd > tv[NK - 1]) {         // guarded sorted insertion
                    float v = nd; int id = gcol;
                    #pragma unroll
                    for (int q = 0; q < NK; ++q) {
                        bool gt = v > tv[q];
                        float ov = tv[q]; int oi = ti[q];
                        tv[q] = gt ? v : ov;  ti[q] = gt ? id : oi;
                        v     = gt ? ov : v;  id    = gt ? oi : id;
                    }
                }
            }
            asm volatile("" ::: "memory");
        }
        __syncthreads();                       // buffer consumed; safe to refill
    }

    // merge the two per-row half-lists (lane L with lane L^16) — wave32 shuffle
    #pragma unroll
    for (int i = 0; i < NK; ++i) {
        float pv = __shfl_xor(tv[i], 16, 32);
        int   pi = __shfl_xor(ti[i], 16, 32);
        if (pv > tv[NK - 1]) {
            float v = pv; int id = pi;
            #pragma unroll
            for (int q = 0; q < NK; ++q) {
                bool gt = v > tv[q];
                float ov = tv[q]; int oi = ti[q];
                tv[q] = gt ? v : ov;  ti[q] = gt ? id : oi;
                v     = gt ? ov : v;  id    = gt ? oi : id;
            }
        }
    }
    if (hs == 0) {
        int* dst = knn_idx + ((size_t)b * NP + myrow) * NK;
        #pragma unroll
        for (int i = 0; i < NK; ++i) dst[i] = ti[i];
    }
}

// ---------------------------------------------------------------------------
// Kernel 3: edge-conv GEMM (K=16 neighbors == WMMA M=16) + BN + LeakyReLU +
// max over K, fused.  y = nbr@W1^T + ctr@(W2-W1)^T.  One wave per point.
// ---------------------------------------------------------------------------
__global__ __launch_bounds__(128) void k_edgeconv(const float* __restrict__ xt,
                                                  const int* __restrict__ knn_idx,
                                                  const float* __restrict__ conv_w,
                                                  const float* __restrict__ bn_gamma,
                                                  const float* __restrict__ bn_beta,
                                                  const float* __restrict__ bn_mean,
                                                  const float* __restrict__ bn_var,
                                                  float* __restrict__ out)
{
    __shared__ float2 ldsW1[32 * 128];      // W1 as [kpair][o]  (32 KB)
    __shared__ float  ldsNbr[4][16 * NC];   // per-wave gathered neighbor rows (16 KB)
    __shared__ float  ldsCtr[4][NC];        // per-wave center row
    __shared__ float  ldsT[4][NO];          // per-wave center term t[o]

    int tid  = threadIdx.x;
    int wave = tid >> 5;
    int lane = tid & 31;
    int hs   = lane >> 4;
    int l16  = lane & 15;

    // stage W1 (first NC columns of conv_w rows) pair-interleaved, once per WG
    #pragma unroll
    for (int i = 0; i < 32; ++i) {
        int flat = tid + i * 128;           // 0..4095 float2
        int o  = flat >> 5;
        int kp = flat & 31;
        float2 v = ((const float2*)(conv_w + (size_t)o * (2 * NC)))[kp];
        ldsW1[kp * 128 + o] = v;
    }
    __syncthreads();

    int p = blockIdx.x * 4 + wave;          // global point id
    int b = p / NP;
    int n = p % NP;
    const float* xtb = xt + (size_t)b * NP * NC;

    // gather the 16 neighbor rows (2 lanes per row) + the center row into LDS
    {
        int r    = lane >> 1;
        int half = lane & 1;
        int nid  = knn_idx[(size_t)p * NK + r];
        const float4* src = (const float4*)(xtb + (size_t)nid * NC + half * 32);
        float4* dst = (float4*)(&ldsNbr[wave][r * NC + half * 32]);
        #pragma unroll
        for (int j = 0; j < 8; ++j) dst[j] = src[j];
        float2 cv = ((const float2*)(xtb + (size_t)n * NC))[lane];
        *(float2*)(&ldsCtr[wave][lane * 2]) = cv;
    }
    asm volatile("" ::: "memory");

    // center term: t[o] = sum_c ctr[c] * (W2[o][c] - W1[o][c]); 4 o's per lane
    #pragma unroll
    for (int j = 0; j < 4; ++j) {
        int o = lane * 4 + j;
        const float* wrow = conv_w + (size_t)o * (2 * NC);
        float t = 0.f;
        for (int c = 0; c < NC; ++c)
            t += ldsCtr[wave][c] * (wrow[NC + c] - wrow[c]);
        ldsT[wave][o] = t;
    }
    asm volatile("" ::: "memory");

    // A fragments: 16 neighbor rows (M = neighbor index k) x 64 features
    v2f afrag[16];
    #pragma unroll
    for (int kk = 0; kk < 16; ++kk) {
        int k = kk * 4 + hs * 2;
        afrag[kk].x = ldsNbr[wave][l16 * NC + k];
        afrag[kk].y = ldsNbr[wave][l16 * NC + k + 1];
    }

    // 8 output tiles of 16 channels; fused BN + LeakyReLU + max over K
    for (int ot = 0; ot < 8; ++ot) {
        int o = ot * 16 + l16;
        v8f acc = {};
        #pragma unroll
        for (int kk = 0; kk < 16; ++kk) {
            float2 bp = ldsW1[(kk * 2 + hs) * 128 + o];
            v2f bf; bf.x = bp.x; bf.y = bp.y;
            acc = __builtin_amdgcn_wmma_f32_16x16x4_f32(
                    false, afrag[kk], false, bf, (short)0, acc, false, false);
        }
        float inv  = bn_gamma[o] / sqrtf(bn_var[o] + BN_EPSF);
        float bias = bn_beta[o] - bn_mean[o] * inv;
        float tval = ldsT[wave][o];
        float m = -__builtin_inff();
        #pragma unroll
        for (int i = 0; i < 8; ++i) {
            float y = (acc[i] + tval) * inv + bias;
            y = (y >= 0.f) ? y : SLOPE * y;
            m = fmaxf(m, y);
        }
        m = fmaxf(m, __shfl_xor(m, 16, 32));   // combine the two 8-row halves
        if (hs == 0)
            out[((size_t)b * NO + o) * NP + n] = m;
    }
}

// ---------------------------------------------------------------------------
extern "C" void kernel_launch(void* const* d_in, const int* in_sizes, int n_in,
                              void* d_out, int out_size, void* d_ws, size_t ws_size,
                              hipStream_t stream)
{
    const float* x      = (const float*)d_in[0];
    const float* conv_w = (const float*)d_in[1];
    const float* gamma  = (const float*)d_in[2];
    const float* beta   = (const float*)d_in[3];
    const float* mean   = (const float*)d_in[4];
    const float* var    = (const float*)d_in[5];
    float* out = (float*)d_out;

    char* ws = (char*)d_ws;
    float* xt = (float*)ws;                                             // 8 MB
    float* sq = (float*)(ws + (size_t)NB * NP * NC * sizeof(float));    // 128 KB
    int*  knn = (int*)(ws + (size_t)NB * NP * NC * sizeof(float)
                          + (size_t)NB * NP * sizeof(float));           // 2 MB

    k_transpose_sq<<<(NB * NP) / 256, 256, 0, stream>>>(x, xt, sq);
    k_knn<<<NB * (NP / 128), 256, 0, stream>>>(xt, sq, knn);
    k_edgeconv<<<(NB * NP) / 4, 128, 0, stream>>>(xt, knn, conv_w,
                                                  gamma, beta, mean, var, out);
}